// CoordinateLayer_23373212025148
// MI455X (gfx1250) — compile-verified
//
#include <hip/hip_runtime.h>
#include <hip/hip_bf16.h>

// RGN coordinate layer on gfx1250:
// coords_i = translation( T_1 * T_2 * ... * T_i ), T_i = closed-form SRF
// transform from dihedral i. Hierarchical parallel scan of 4x4 matrix
// products; combine op = V_WMMA_F32_16X16X4_F32 packing four independent
// 4x4 compositions per wave-instruction.

#define L_STEPS 4096
#define NDIH    3
#define BATCH   32
#define CHAIN   (L_STEPS * NDIH)   // 12288 points per chain
#define CHUNK   256                // elements scanned per workgroup
#define NCHUNK  (CHAIN / CHUNK)    // 48 chunks per chain
#define MST     17                 // LDS matrix stride in floats (17 coprime 64 banks)
#define PI_F    3.14159265358979f

typedef __attribute__((ext_vector_type(2))) float v2f;
typedef __attribute__((ext_vector_type(8))) float v8f;

// ---------------------------------------------------------------------------
// Compose 4 pairs of 4x4 row-major matrices with one V_WMMA_F32_16X16X4_F32.
//   slots s0..s0+3:  dst[s] = src[s-off] * src[s]   (identity if s < off)
// A (16x4)  = [M0;M1;M2;M3] stacked  -> lanes 0-15 hold K0,K1; 16-31 hold K2,K3
// B (4x16)  = [N0 N1 N2 N3] side-by-side, same K split across wave halves
// D (16x16) diagonal 4x4 blocks = M_g * N_g.
// Called by a full wave (EXEC all ones) -- no divergence before the WMMA.
// ---------------------------------------------------------------------------
__device__ __forceinline__ void wmma_compose4(const float* __restrict__ src,
                                              const float* __restrict__ idm,
                                              float* __restrict__ dst,
                                              int s0, int off, int lane)
{
    const int half = lane >> 4;        // 0: K=0,1   1: K=2,3
    const int Ml   = lane & 15;        // A row / B column within 16
    const int g    = Ml >> 2;          // which 4x4 block (0..3)
    const int r    = Ml & 3;           // row within block
    const int c    = Ml & 3;           // col within block (B side)
    const int k0   = half ? 2 : 0;

    // A element: matrix M_g = src[s0+g - off] (or identity)
    const int    sg = s0 + g;
    const float* mm = (sg >= off) ? (src + (size_t)(sg - off) * MST) : idm;
    v2f a;
    a.x = mm[r * 4 + k0];
    a.y = mm[r * 4 + k0 + 1];

    // B element: matrix N_g = src[s0+g]
    const float* nn = src + (size_t)(s0 + g) * MST;
    v2f bb;
    bb.x = nn[k0 * 4 + c];
    bb.y = nn[(k0 + 1) * 4 + c];

    v8f cz = {0.f, 0.f, 0.f, 0.f, 0.f, 0.f, 0.f, 0.f};
    // 8 args: (neg_a, A, neg_b, B, c_mod, C, reuse_a, reuse_b)
    v8f d = __builtin_amdgcn_wmma_f32_16x16x4_f32(false, a, false, bb,
                                                  (short)0, cz, false, false);

    // Scatter diagonal blocks. Vgpr element rr of this lane holds
    //   D[M][N],  M = rr + 8*half,  N = lane&15.
    // Keep iff it lies in a diagonal 4x4 block: (M>>2) == (N>>2).
    // rr is a compile-time constant after unroll -> direct register extract,
    // plain per-lane predicated ds_store (no cndmask ladders).
#pragma unroll
    for (int rr = 0; rr < 8; ++rr) {
        const int mb = (rr >> 2) + (half << 1);      // block row of this element
        if (mb == g) {
            dst[(size_t)(s0 + mb) * MST + (rr & 3) * 4 + c] = d[rr];
        }
    }
}

// ---------------------------------------------------------------------------
// Kernel 1: build local transforms + in-chunk inclusive scan (WMMA tree).
// grid = (NCHUNK, BATCH), block = 256 (8 wave32)
// ---------------------------------------------------------------------------
__global__ void __launch_bounds__(256)
rgn_local_scan(const float* __restrict__ dih_in,
               float* __restrict__ tloc,     // [CHAIN][BATCH][3] local translations
               float* __restrict__ agg)      // [BATCH][NCHUNK][16] chunk products
{
    __shared__ float S0[CHUNK * MST];
    __shared__ float S1[CHUNK * MST];
    __shared__ float IDM[16];

    const int tid   = threadIdx.x;
    const int lane  = tid & 31;
    const int wave  = tid >> 5;
    const int chunk = blockIdx.x;
    const int b     = blockIdx.y;

    if (tid < 16) IDM[tid] = (tid % 5 == 0) ? 1.f : 0.f;   // diag 0,5,10,15

    // ---- build T_i (SRF local transform) for element gi = chunk*CHUNK+tid ----
    const int gi = chunk * CHUNK + tid;
    const int l  = gi / NDIH;
    const int j  = gi - l * NDIH;

    const float BL[3] = {145.801f, 152.326f, 132.868f};
    const float BA[3] = {2.124f, 1.941f, 2.028f};
    const float bl    = BL[j];
    const float alpha = PI_F - BA[j];
    const float rc    = bl * cosf(alpha);
    const float rs    = bl * sinf(alpha);

    const float dih = dih_in[((size_t)l * BATCH + b) * NDIH + j];
    float sd, cd;
    sincosf(dih, &sd, &cd);
    const float px = rc, py = cd * rs, pz = sd * rs;

    // u = p/|p| (|p| == bond length exactly); w = normalize(e_x x u); v = w x u
    const float inv_r = 1.0f / bl;
    const float ux = px * inv_r, uy = py * inv_r, uz = pz * inv_r;
    const float inv_s = rsqrtf(fmaxf(py * py + pz * pz, 1e-12f));
    const float wx = 0.f, wy = -pz * inv_s, wz = py * inv_s;
    const float vx = wy * uz - wz * uy;
    const float vy = wz * ux - wx * uz;
    const float vz = wx * uy - wy * ux;

    float* T = S0 + (size_t)tid * MST;     // row-major 4x4, columns = [u v w t]
    T[0] = ux; T[1] = vx; T[2]  = wx; T[3]  = px;
    T[4] = uy; T[5] = vy; T[6]  = wy; T[7]  = py;
    T[8] = uz; T[9] = vz; T[10] = wz; T[11] = pz;
    T[12] = 0.f; T[13] = 0.f; T[14] = 0.f; T[15] = 1.f;
    __syncthreads();

    // ---- Kogge-Stone inclusive scan, combine = left * right via WMMA ----
#pragma unroll
    for (int rnd = 0; rnd < 8; ++rnd) {          // off = 1..128
        const int off = 1 << rnd;
        float* src = (rnd & 1) ? S1 : S0;
        float* dst = (rnd & 1) ? S0 : S1;
#pragma unroll
        for (int it = 0; it < CHUNK / 32; ++it) {
            wmma_compose4(src, IDM, dst, it * 32 + wave * 4, off, lane);
        }
        __syncthreads();
    }
    // 8 rounds end back in S0.

    // ---- emit local translation (all kernel 3 needs) + chunk aggregate ----
    const float* fin = S0 + (size_t)tid * MST;
    const size_t oi = ((size_t)gi * BATCH + b) * 3;
    tloc[oi + 0] = fin[3];
    tloc[oi + 1] = fin[7];
    tloc[oi + 2] = fin[11];
    if (tid < 16)
        agg[((size_t)b * NCHUNK + chunk) * 16 + tid] = S0[(CHUNK - 1) * MST + tid];
}

// ---------------------------------------------------------------------------
// Kernel 2: exclusive scan of the 48 chunk aggregates per chain (serial, tiny).
// block = 32 threads, one chain each.
// ---------------------------------------------------------------------------
__global__ void rgn_agg_scan(const float* __restrict__ agg,
                             float* __restrict__ pref)
{
    const int b = threadIdx.x;
    if (b >= BATCH) return;
    float P[16] = {1.f, 0.f, 0.f, 0.f,
                   0.f, 1.f, 0.f, 0.f,
                   0.f, 0.f, 1.f, 0.f,
                   0.f, 0.f, 0.f, 1.f};
    for (int k = 0; k < NCHUNK; ++k) {
        const size_t base = ((size_t)b * NCHUNK + k) * 16;
#pragma unroll
        for (int e = 0; e < 16; ++e) pref[base + e] = P[e];   // exclusive prefix
        const float* A = agg + base;
        float T[16];
#pragma unroll
        for (int r = 0; r < 4; ++r)
#pragma unroll
            for (int c = 0; c < 4; ++c)
                T[r * 4 + c] = fmaf(P[r * 4 + 0], A[0 * 4 + c],
                               fmaf(P[r * 4 + 1], A[1 * 4 + c],
                               fmaf(P[r * 4 + 2], A[2 * 4 + c],
                                    P[r * 4 + 3] * A[3 * 4 + c])));
#pragma unroll
        for (int e = 0; e < 16; ++e) P[e] = T[e];
    }
}

// ---------------------------------------------------------------------------
// Kernel 3: coord = P_rot * t_local + P_t   (3x4 affine apply per point)
// ---------------------------------------------------------------------------
__global__ void __launch_bounds__(256)
rgn_apply(const float* __restrict__ tloc,
          const float* __restrict__ pref,
          float* __restrict__ out)
{
    const int t = blockIdx.x * 256 + threadIdx.x;   // t = i*BATCH + b
    if (t >= CHAIN * BATCH) return;
    const int i = t / BATCH;
    const int b = t - i * BATCH;
    const int chunk = i / CHUNK;
    const float* P = pref + ((size_t)b * NCHUNK + chunk) * 16;
    const size_t oi = (size_t)t * 3;
    const float tx = tloc[oi + 0], ty = tloc[oi + 1], tz = tloc[oi + 2];
    out[oi + 0] = fmaf(P[0], tx, fmaf(P[1], ty, fmaf(P[2],  tz, P[3])));
    out[oi + 1] = fmaf(P[4], tx, fmaf(P[5], ty, fmaf(P[6],  tz, P[7])));
    out[oi + 2] = fmaf(P[8], tx, fmaf(P[9], ty, fmaf(P[10], tz, P[11])));
}

// ---------------------------------------------------------------------------
extern "C" void kernel_launch(void* const* d_in, const int* in_sizes, int n_in,
                              void* d_out, int out_size, void* d_ws, size_t ws_size,
                              hipStream_t stream)
{
    (void)in_sizes; (void)n_in; (void)out_size; (void)ws_size;
    const float* dih = (const float*)d_in[0];
    float* out = (float*)d_out;

    float* ws   = (float*)d_ws;
    float* tloc = ws;                                        // CHAIN*BATCH*3
    float* agg  = tloc + (size_t)CHAIN * BATCH * 3;          // BATCH*NCHUNK*16
    float* pref = agg + (size_t)BATCH * NCHUNK * 16;         // BATCH*NCHUNK*16

    dim3 gridB(NCHUNK, BATCH, 1);
    rgn_local_scan<<<gridB, 256, 0, stream>>>(dih, tloc, agg);
    rgn_agg_scan<<<1, 32, 0, stream>>>(agg, pref);
    const int total = CHAIN * BATCH;
    rgn_apply<<<(total + 255) / 256, 256, 0, stream>>>(tloc, pref, out);
}